// GCN_22608707846476
// MI455X (gfx1250) — compile-verified
//
#include <hip/hip_runtime.h>

typedef __attribute__((ext_vector_type(2))) float v2f;
typedef __attribute__((ext_vector_type(8))) float v8f;

// ---------------- small utility kernels ----------------

__global__ void k_zero_int(int* __restrict__ p, int n) {
    int i = blockIdx.x * blockDim.x + threadIdx.x;
    if (i < n) p[i] = 0;
}

// count in-degree (excluding self-loop) per destination node
__global__ void k_count(const long long* __restrict__ dst,
                        int* __restrict__ cnt, int E) {
    int i = blockIdx.x * blockDim.x + threadIdx.x;
    if (i < E) atomicAdd(&cnt[(int)dst[i]], 1);
}

// dinv[n] = rsqrt(deg) with deg = cnt[n] + 1 (self-loop)
__global__ void k_dinv(const int* __restrict__ cnt,
                       float* __restrict__ dinv, int N) {
    int i = blockIdx.x * blockDim.x + threadIdx.x;
    if (i < N) dinv[i] = rsqrtf((float)cnt[i] + 1.0f);
}

// single-block exclusive scan of cnt[0..N) -> offs[0..N], LDS Hillis-Steele
#define SCAN_T 1024
__global__ void k_scan_offsets(const int* __restrict__ cnt,
                               int* __restrict__ offs, int N) {
    __shared__ int sums[SCAN_T];
    const int t = threadIdx.x;
    const int chunk = (N + SCAN_T - 1) / SCAN_T;
    const int lo = t * chunk;
    const int hi = (lo + chunk < N) ? lo + chunk : N;
    int s = 0;
    for (int i = lo; i < hi; ++i) s += cnt[i];
    sums[t] = s;
    __syncthreads();
    for (int off = 1; off < SCAN_T; off <<= 1) {
        int v = (t >= off) ? sums[t - off] : 0;
        __syncthreads();
        if (t >= off) sums[t] += v;
        __syncthreads();
    }
    int run = (t > 0) ? sums[t - 1] : 0;   // exclusive base of this chunk
    for (int i = lo; i < hi; ++i) { offs[i] = run; run += cnt[i]; }
    if (t == SCAN_T - 1) offs[N] = run;    // == total E
}

// bucket edges into CSR slots as packed {src:int32, norm:float}
__global__ void k_fill(const long long* __restrict__ src,
                       const long long* __restrict__ dst,
                       const float* __restrict__ dinv,
                       const int* __restrict__ offs,
                       int* __restrict__ cursor,
                       int2* __restrict__ eb, int E) {
    int i = blockIdx.x * blockDim.x + threadIdx.x;
    if (i >= E) return;
    const int s = (int)src[i], d = (int)dst[i];
    const int pos = offs[d] + atomicAdd(&cursor[d], 1);
    int2 ent;
    ent.x = s;
    ent.y = __float_as_int(dinv[s] * dinv[d]);
    eb[pos] = ent;
}

// ---------------- dense GEMM: V_WMMA_F32_16X16X4_F32, B staged in LDS -------
// D[M x 64] = act(A)[M x K] @ B[K x 64];  block = 128 thr = 4 waves,
// wave w owns the 16x16 tile at (16*blockIdx.x, 16*w). RELU is a template
// parameter so the non-relu instantiation has a clean WMMA inner loop.
template <int RELU>
__global__ void k_gemm_wmma_f32(const float* __restrict__ A,
                                const float* __restrict__ B,
                                float* __restrict__ D,
                                int M, int K) {
    extern __shared__ float Bs[];                 // K*64 floats
    // cooperative B load (whole B matrix: 16-32 KB, read once per block)
    {
        const float4* B4  = (const float4*)B;
        float4*       Bs4 = (float4*)Bs;
        const int nB4 = K * 16;                   // K*64/4
        for (int i = threadIdx.x; i < nB4; i += 128) Bs4[i] = B4[i];
    }
    __syncthreads();

    const int lane = threadIdx.x & 31;
    const int wave = threadIdx.x >> 5;
    const int m0   = blockIdx.x * 16;
    const int n0   = wave * 16;
    const int lm   = lane & 15;                   // A row / B,D col within tile
    const int hb   = lane >> 4;                   // half-wave selector

    int arow = m0 + lm;
    if (arow >= M) arow = M - 1;                  // clamp: EXEC must stay all-1s
    const float* Ar = A + (size_t)arow * (size_t)K;

    v8f c = {};
    for (int k0 = 0; k0 < K; k0 += 4) {
        const int ka = k0 + hb * 2;               // lanes 0-15: K=k0,k0+1; 16-31: +2,+3
        v2f a, b;
        a.x = Ar[ka];
        a.y = Ar[ka + 1];
        if (RELU) { a.x = fmaxf(a.x, 0.0f); a.y = fmaxf(a.y, 0.0f); }
        b.x = Bs[ka * 64 + n0 + lm];
        b.y = Bs[(ka + 1) * 64 + n0 + lm];
        c = __builtin_amdgcn_wmma_f32_16x16x4_f32(
                false, a, false, b, (short)0, c, false, false);
    }

#pragma unroll
    for (int r = 0; r < 8; ++r) {
        const int orow = m0 + hb * 8 + r;         // C/D layout: M = 8*hb + r
        if (orow < M) D[(size_t)orow * 64 + n0 + lm] = c[r];
    }
}

// ---------------- CSR gather aggregation (no atomics) ----------------
// one wave per node; lane owns features [2*lane, 2*lane+1].
// agg[n,:] = bias + dinv[n]^2 * h[n,:] + sum_{(s,w) in CSR[n]} w * h[s,:]
__global__ void k_aggregate(const float* __restrict__ h,
                            const int2* __restrict__ eb,
                            const int* __restrict__ offs,
                            const float* __restrict__ dinv,
                            const float* __restrict__ bias,
                            float* __restrict__ agg, int N) {
    const int n = (int)((blockIdx.x * blockDim.x + threadIdx.x) >> 5);
    if (n >= N) return;
    const int lane = threadIdx.x & 31;
    const float2* h2 = (const float2*)h;

    const float di = dinv[n];
    float2 hv = h2[(size_t)n * 32 + lane];
    float accx = hv.x * di * di;                  // self-loop term
    float accy = hv.y * di * di;

    const int beg = offs[n], end = offs[n + 1];
    for (int i = beg; i < end; ++i) {
        const int2 e = eb[i];                     // broadcast 8B load
        const float w = __int_as_float(e.y);
        const float2 v = h2[(size_t)e.x * 32 + lane];  // coalesced 256B gather
        accx += v.x * w;
        accy += v.y * w;
    }
    float2 o;
    o.x = accx + bias[2 * lane];
    o.y = accy + bias[2 * lane + 1];
    ((float2*)agg)[(size_t)n * 32 + lane] = o;
}

// ---------------- final projection ----------------
// out[n,t] = bl[t] + sum_f relu(agg[n,f]) * Wl[f,t]
__global__ void k_final(const float* __restrict__ agg,
                        const float* __restrict__ Wl,
                        const float* __restrict__ bl,
                        float* __restrict__ out, int N, int T) {
    int n = blockIdx.x * blockDim.x + threadIdx.x;
    if (n >= N) return;
    for (int t = 0; t < T; ++t) {
        float acc = bl[t];
#pragma unroll 8
        for (int f = 0; f < 64; ++f)
            acc += fmaxf(agg[(size_t)n * 64 + f], 0.0f) * Wl[f * T + t];
        out[(size_t)n * T + t] = acc;
    }
}

// ---------------- launcher ----------------

extern "C" void kernel_launch(void* const* d_in, const int* in_sizes, int n_in,
                              void* d_out, int out_size, void* d_ws, size_t ws_size,
                              hipStream_t stream) {
    const float*     x   = (const float*)d_in[0];
    const long long* ei  = (const long long*)d_in[1];   // int64 edge_index [2, E]
    const float*     W1  = (const float*)d_in[2];
    const float*     b1  = (const float*)d_in[3];
    const float*     W2  = (const float*)d_in[4];
    const float*     b2  = (const float*)d_in[5];
    const float*     Wl  = (const float*)d_in[6];
    const float*     bl  = (const float*)d_in[7];
    float*           out = (float*)d_out;

    const int F = 128;
    const int H = 64;
    const int N = in_sizes[0] / F;        // 100000
    const int E = in_sizes[1] / 2;        // 1600000
    const int T = in_sizes[6] / H;        // 1

    const long long* src = ei;            // row 0
    const long long* dst = ei + E;        // row 1

    // ---- workspace carving (~66 MB, L2-resident) ----
    char* w = (char*)d_ws;
    auto take = [&](size_t bytes) {
        char* p = w;
        w += (bytes + 255) & ~(size_t)255;
        return (void*)p;
    };
    float* dinv   = (float*)take((size_t)N * 4);
    int*   cnt    = (int*)  take((size_t)N * 4);        // counts, then cursor
    int*   offs   = (int*)  take(((size_t)N + 1) * 4);
    int2*  eb     = (int2*) take((size_t)E * 8);        // CSR {src, norm}
    float* h      = (float*)take((size_t)N * H * 4);
    float* agg    = (float*)take((size_t)N * H * 4);

    const int TB = 256;
    const int gN   = (N + TB - 1) / TB;
    const int gE   = (E + TB - 1) / TB;
    const int gM16 = (N + 15) / 16;
    const int gW   = (int)(((size_t)N * 32 + TB - 1) / TB);  // 1 wave per node

    // ---- build CSR + normalization (once; reused by both layers) ----
    k_zero_int<<<gN, TB, 0, stream>>>(cnt, N);
    k_count<<<gE, TB, 0, stream>>>(dst, cnt, E);
    k_scan_offsets<<<1, SCAN_T, 0, stream>>>(cnt, offs, N);
    k_dinv<<<gN, TB, 0, stream>>>(cnt, dinv, N);
    k_zero_int<<<gN, TB, 0, stream>>>(cnt, N);               // reuse as cursor
    k_fill<<<gE, TB, 0, stream>>>(src, dst, dinv, offs, cnt, eb, E);

    // ---- layer 1: h = x @ W1 ; agg = b1 + norm-weighted gather ----
    k_gemm_wmma_f32<0><<<gM16, 128, (size_t)F * 64 * 4, stream>>>(x, W1, h, N, F);
    k_aggregate<<<gW, TB, 0, stream>>>(h, eb, offs, dinv, b1, agg, N);

    // ---- layer 2: h = relu(agg) @ W2 ; agg = b2 + gather ----
    k_gemm_wmma_f32<1><<<gM16, 128, (size_t)H * 64 * 4, stream>>>(agg, W2, h, N, H);
    k_aggregate<<<gW, TB, 0, stream>>>(h, eb, offs, dinv, b2, agg, N);

    // ---- final projection (relu folded into load) ----
    k_final<<<gN, TB, 0, stream>>>(agg, Wl, bl, out, N, T);
}